// FeatureLayer_79723182948737
// MI455X (gfx1250) — compile-verified
//
#include <hip/hip_runtime.h>
#include <stdint.h>

// Problem constants from the reference:
//   B=1, N=1000, T=256, M=258, D=512, NUMERIC=10
//   out row = 6*512 + 10 = 3082 floats = 1541 float2 (8B aligned rows)
#define N_CAND 1000
#define T_DIM  256
#define M_DIM  258
#define D_DIM  512
#define ROW_F2 1541   // 3082 floats / 2

typedef __attribute__((ext_vector_type(2))) float f2;

__global__ __launch_bounds__(256)
void FeatureGather_async(const int*   __restrict__ cand,  // (N,6)
                         const float* __restrict__ num,   // (N,10)
                         const float* __restrict__ blo,   // (N,256,512)
                         const float* __restrict__ att,   // (N,258,512)
                         float*       __restrict__ out)   // (N,3082)
{
    // 6 float2 slots per thread, wave-private -> only s_wait_asynccnt needed,
    // no workgroup barrier.
    __shared__ f2 lbuf[256 * 6];

    const int n = blockIdx.x;
    const int t = threadIdx.x;

    const int c0 = cand[n * 6 + 0];
    const int c1 = cand[n * 6 + 1];
    const int c2 = cand[n * 6 + 2];
    const int c3 = cand[n * 6 + 3];
    const int c4 = cand[n * 6 + 4];
    const int c5 = cand[n * 6 + 5];

    const f2* blo2 = (const f2*)blo;
    const f2* att2 = (const f2*)att;

    // Source rows in concat order: g1, g2, g4, g5, parent, child
    const f2* srows[6];
    srows[0] = blo2 + ((size_t)n * T_DIM + (size_t)c1) * (D_DIM / 2);
    srows[1] = blo2 + ((size_t)n * T_DIM + (size_t)c2) * (D_DIM / 2);
    srows[2] = blo2 + ((size_t)n * T_DIM + (size_t)c4) * (D_DIM / 2);
    srows[3] = blo2 + ((size_t)n * T_DIM + (size_t)c5) * (D_DIM / 2);
    srows[4] = att2 + ((size_t)n * M_DIM + (size_t)(c0 + 2)) * (D_DIM / 2);
    srows[5] = att2 + ((size_t)n * M_DIM + (size_t)(c3 + 2)) * (D_DIM / 2);

    f2* out2 = (f2*)out + (size_t)n * ROW_F2;

    // Low 32 bits of a flat pointer into __shared__ == wave-relative LDS byte
    // offset (ISA 10.2: LDS aperture address maps via addr[31:0]).
    const uint32_t lds_base = (uint32_t)(size_t)(void*)(&lbuf[t * 6]);

    // Phase 1: async DMA gather  global -> LDS  (ASYNCcnt)
#pragma unroll
    for (int s = 0; s < 6; ++s) {
        const uint64_t gsrc = (uint64_t)(size_t)(srows[s] + t);
        const uint32_t laddr = lds_base + (uint32_t)(s * 8);
        asm volatile("global_load_async_to_lds_b64 %0, %1, off"
                     :
                     : "v"(laddr), "v"(gsrc)
                     : "memory");
    }

    // Wait for this wave's async loads to land in LDS (wave-private slots,
    // so no cross-wave barrier is required).
    asm volatile("s_wait_asynccnt 0" ::: "memory");

    // Phase 2: async DMA scatter  LDS -> global  (ASYNCcnt; S_ENDPGM's
    // implicit wait-idle guarantees completion before the kernel retires).
#pragma unroll
    for (int s = 0; s < 6; ++s) {
        const uint64_t gdst = (uint64_t)(size_t)(out2 + s * (D_DIM / 2) + t);
        const uint32_t laddr = lds_base + (uint32_t)(s * 8);
        asm volatile("global_store_async_from_lds_b64 %0, %1, off"
                     :
                     : "v"(gdst), "v"(laddr)
                     : "memory");
    }

    // Numeric tail: 10 floats = 5 float2 per row (8B aligned), threads 0..4.
    if (t < 5) {
        const f2 v = ((const f2*)num)[(size_t)n * 5 + t];
        __builtin_nontemporal_store(v, &out2[6 * (D_DIM / 2) + t]);
    }
}

extern "C" void kernel_launch(void* const* d_in, const int* in_sizes, int n_in,
                              void* d_out, int out_size, void* d_ws, size_t ws_size,
                              hipStream_t stream) {
    const int*   cand = (const int*)  d_in[0]; // candidates            (1,1000,6)   int32
    const float* num  = (const float*)d_in[1]; // numeric features      (1,1000,10)  f32
    const float* blo  = (const float*)d_in[2]; // stacked_bi_lstm_output(1,1000,256,512) f32
    const float* att  = (const float*)d_in[3]; // stacked_attended_nodes(1,1000,258,512) f32
    float* out = (float*)d_out;                // (1,1000,3082) f32

    FeatureGather_async<<<N_CAND, 256, 0, stream>>>(cand, num, blo, att, out);
}